// GCN_20615843021391
// MI455X (gfx1250) — compile-verified
//
#include <hip/hip_runtime.h>
#include <hip/hip_bf16.h>

typedef float v2f __attribute__((ext_vector_type(2)));
typedef float v8f __attribute__((ext_vector_type(8)));

#define D_FEAT 128
#define LDS_PITCH 132   // 128 + 4 pad; 132 floats = 528 B = 33*16 B -> float4-aligned rows,
                        // and column reads from 16 consecutive rows hit distinct banks.

// ---------------------------------------------------------------------------
// Kernel 1: zero the workspace (agg[N*128] ++ deg[N]), float4-wide
// ---------------------------------------------------------------------------
__global__ __launch_bounds__(256) void gcn_zero_kernel(float4* __restrict__ p4, int n4) {
    int i = blockIdx.x * blockDim.x + threadIdx.x;
    if (i < n4) p4[i] = make_float4(0.f, 0.f, 0.f, 0.f);
}

// ---------------------------------------------------------------------------
// Kernel 2: edge scatter. One 32-lane wave per edge; lane q handles dims
// [4q, 4q+3] via a float4 gather of the source row + 4 f32 hardware atomics
// into agg. Lane 0 also bumps the float degree counter.
// ---------------------------------------------------------------------------
__global__ __launch_bounds__(256) void gcn_scatter_kernel(
    const float4* __restrict__ feat4,   // feature viewed as [N][32] float4
    const int*    __restrict__ src,
    const int*    __restrict__ dst,
    float*        __restrict__ agg,     // [N][128]
    float*        __restrict__ deg,     // [N]
    int n_edges)
{
    int idx  = blockIdx.x * blockDim.x + threadIdx.x;
    int e    = idx >> 5;
    if (e >= n_edges) return;
    int lane = idx & 31;

    int s = src[e];
    int d = dst[e];

    float4 v = feat4[(size_t)s * 32 + lane];
    float* base = agg + (size_t)d * D_FEAT + lane * 4;
    unsafeAtomicAdd(base + 0, v.x);
    unsafeAtomicAdd(base + 1, v.y);
    unsafeAtomicAdd(base + 2, v.z);
    unsafeAtomicAdd(base + 3, v.w);
    if (lane == 0) unsafeAtomicAdd(deg + d, 1.0f);
}

// ---------------------------------------------------------------------------
// Kernel 3: apply phase  out = relu(h @ W + b),  h = deg>0 ? agg/deg : feature
//
// One block (8 waves) per 32-node row tile. Wave w owns output column tile w
// (16 cols) and computes TWO 16x16 M-tiles with independent accumulators:
// each K-step loads the B fragment once and feeds two independent
// v_wmma_f32_16x16x4_f32 ops -> halves W traffic and breaks the WMMA
// latency chain.
//
// ISA layouts (CDNA5 ISA):
//   A (16x4 f32, 2 VGPR): lanes 0-15 -> M=lane,    K={0,1}
//                         lanes16-31 -> M=lane-16, K={2,3}
//   B ( 4x16 f32, 2 VGPR): same K striping, N = lane&15
//   C/D (16x16 f32, 8 VGPR): VGPR g -> M = g + 8*(lane>=16), N = lane&15
// ---------------------------------------------------------------------------
__global__ __launch_bounds__(256) void gcn_gemm_kernel(
    const float* __restrict__ agg,
    const float* __restrict__ deg,
    const float* __restrict__ feat,
    const float* __restrict__ W,      // [128][128] row-major (k, n)
    const float* __restrict__ bias,   // [128]
    float*       __restrict__ out,    // [N][128]
    int n_nodes)
{
    __shared__ float hlds[32 * LDS_PITCH];   // 32 rows x 128 (+pad) = 16.9 KB

    const int node0 = blockIdx.x * 32;
    const int t     = threadIdx.x;

    // ---- stage h tile (32 x 128) into LDS with float4 loads/stores ----
    #pragma unroll
    for (int i = 0; i < 4; ++i) {
        int idx  = t + i * 256;          // 0..1023 float4 slots
        int row  = idx >> 5;             // 0..31
        int c4   = idx & 31;             // float4 column
        int node = node0 + row;
        float4 h4 = make_float4(0.f, 0.f, 0.f, 0.f);
        if (node < n_nodes) {
            float  dg  = deg[node];
            float4 a4  = ((const float4*)agg )[(size_t)node * 32 + c4];
            float4 f4  = ((const float4*)feat)[(size_t)node * 32 + c4];
            float  inv = 1.0f / fmaxf(dg, 1.0f);
            bool   has = dg > 0.0f;
            h4.x = has ? a4.x * inv : f4.x;
            h4.y = has ? a4.y * inv : f4.y;
            h4.z = has ? a4.z * inv : f4.z;
            h4.w = has ? a4.w * inv : f4.w;
        }
        *(float4*)&hlds[row * LDS_PITCH + c4 * 4] = h4;   // 16B-aligned
    }
    __syncthreads();

    const int wave  = t >> 5;            // 0..7 -> output column tile
    const int lane  = t & 31;
    const int m     = lane & 15;         // A row / B-D col within tile
    const int khalf = (lane >> 4) * 2;   // 0 for lanes 0-15, 2 for 16-31
    const int ncol  = wave * 16 + m;     // global output column

    v8f acc0 = {};                       // M-tile rows node0+0..15
    v8f acc1 = {};                       // M-tile rows node0+16..31
    #pragma unroll
    for (int k0 = 0; k0 < D_FEAT; k0 += 4) {
        const int ka = k0 + khalf;
        v2f b, a0, a1;
        b.x  = W[(size_t)ka       * D_FEAT + ncol];
        b.y  = W[(size_t)(ka + 1) * D_FEAT + ncol];
        a0.x = hlds[m        * LDS_PITCH + ka];
        a0.y = hlds[m        * LDS_PITCH + ka + 1];
        a1.x = hlds[(m + 16) * LDS_PITCH + ka];
        a1.y = hlds[(m + 16) * LDS_PITCH + ka + 1];
        acc0 = __builtin_amdgcn_wmma_f32_16x16x4_f32(
            false, a0, false, b, (short)0, acc0, false, false);
        acc1 = __builtin_amdgcn_wmma_f32_16x16x4_f32(
            false, a1, false, b, (short)0, acc1, false, false);
    }

    const float bb = bias[ncol];
    #pragma unroll
    for (int g = 0; g < 8; ++g) {
        int row   = g + ((lane >> 4) * 8);
        int nodeA = node0 + row;          // tile 0
        int nodeB = node0 + 16 + row;     // tile 1
        if (nodeA < n_nodes)
            out[(size_t)nodeA * D_FEAT + ncol] = fmaxf(acc0[g] + bb, 0.0f);
        if (nodeB < n_nodes)
            out[(size_t)nodeB * D_FEAT + ncol] = fmaxf(acc1[g] + bb, 0.0f);
    }
}

// ---------------------------------------------------------------------------
// Launcher
// ---------------------------------------------------------------------------
extern "C" void kernel_launch(void* const* d_in, const int* in_sizes, int n_in,
                              void* d_out, int out_size, void* d_ws, size_t ws_size,
                              hipStream_t stream) {
    const float* feature = (const float*)d_in[0];   // [N,128]
    const float* W       = (const float*)d_in[1];   // [128,128]
    const float* b       = (const float*)d_in[2];   // [128]
    const int*   src     = (const int*)d_in[3];     // [E]
    const int*   dst     = (const int*)d_in[4];     // [E]
    float*       out     = (float*)d_out;

    const int n_nodes = in_sizes[0] / D_FEAT;
    const int n_edges = in_sizes[3];

    float* agg = (float*)d_ws;                      // [N*128]
    float* deg = agg + (size_t)n_nodes * D_FEAT;    // [N]

    // 1) zero agg + deg (N*(128+1) floats; N is a multiple of 4 here, and
    //    rounding up the float4 count only touches workspace slack)
    {
        int n  = n_nodes * (D_FEAT + 1);
        int n4 = (n + 3) / 4;
        int blocks = (n4 + 255) / 256;
        gcn_zero_kernel<<<blocks, 256, 0, stream>>>((float4*)d_ws, n4);
    }
    // 2) scatter: one wave per edge
    {
        long long threads = (long long)n_edges * 32;
        int blocks = (int)((threads + 255) / 256);
        gcn_scatter_kernel<<<blocks, 256, 0, stream>>>(
            (const float4*)feature, src, dst, agg, deg, n_edges);
    }
    // 3) apply: dual-tile WMMA GEMM + bias + ReLU
    {
        int blocks = (n_nodes + 31) / 32;
        gcn_gemm_kernel<<<blocks, 256, 0, stream>>>(
            agg, deg, feature, W, b, out, n_nodes);
    }
}